// self_attention_41738492183083
// MI455X (gfx1250) — compile-verified
//
#include <hip/hip_runtime.h>
#include <stdint.h>

typedef float v2f __attribute__((ext_vector_type(2)));
typedef float v8f __attribute__((ext_vector_type(8)));

constexpr int kN  = 8192;   // Z*H*W = 8*32*32
constexpr int kC  = 64;
constexpr int kC8 = 8;
constexpr float kLog2e = 1.44269504088896340736f;

// workspace layout per image (floats)
constexpr int F_OFF = 0;                 // f  [N][8]   row-major
constexpr int G_OFF = kN * kC8;          // g  [8][N]
constexpr int V_OFF = 2 * kN * kC8;      // v  (later v' = gamma*v/Z) [64][N]
constexpr int Z_OFF = V_OFF + kC * kN;   // Z  [N]
constexpr int IMG_STRIDE = Z_OFF + kN;

static __device__ inline v8f wmma_f32(v2f a, v2f b, v8f c) {
  // D = A(16x4 f32) * B(4x16 f32) + C(16x16 f32)
  return __builtin_amdgcn_wmma_f32_16x16x4_f32(false, a, false, b, (short)0, c,
                                               false, false);
}

// Async DMA: copies 16 bytes per lane from global gsrc to LDS lds_dst, tracked by ASYNCcnt.
static __device__ __forceinline__ void async_fill_b128(const float* gsrc,
                                                       float* lds_dst) {
  // generic LDS pointer: addr[31:0] is the wave-relative LDS byte address
  uint32_t laddr = (uint32_t)(uintptr_t)lds_dst;
  asm volatile("global_load_async_to_lds_b128 %0, %1, off"
               :: "v"(laddr), "v"(gsrc) : "memory");
}
static __device__ __forceinline__ void wait_async0() {
  asm volatile("s_wait_asynccnt 0x0" ::: "memory");
}

// ---------------------------------------------------------------------------
// Kernel 1: pointwise 1x1x1 conv projections  f = Wf@x+bf, g = Wg@x+bg,
// v = Wh@x+bh.  One thread per spatial position n; weights staged in LDS.
// grid = 64 blocks x 256 threads (blocks 0..31 -> image 1, 32..63 -> image 2)
// ---------------------------------------------------------------------------
__global__ __launch_bounds__(256) void proj_kernel(
    const float* __restrict__ x1, const float* __restrict__ x2,
    const float* __restrict__ Wf, const float* __restrict__ bf,
    const float* __restrict__ Wg, const float* __restrict__ bg,
    const float* __restrict__ Wh1, const float* __restrict__ bh1,
    const float* __restrict__ Wh2, const float* __restrict__ bh2,
    float* __restrict__ ws)
{
  __shared__ float sWf[kC8 * kC];
  __shared__ float sWg[kC8 * kC];
  __shared__ float sWh[kC * kC];
  __shared__ float sb[2 * kC8 + kC];

  const int img = blockIdx.x >> 5;
  const float* x  = img ? x2  : x1;
  const float* Wh = img ? Wh2 : Wh1;
  const float* bh = img ? bh2 : bh1;

  for (int i = threadIdx.x; i < kC8 * kC; i += 256) { sWf[i] = Wf[i]; sWg[i] = Wg[i]; }
  for (int i = threadIdx.x; i < kC * kC; i += 256) sWh[i] = Wh[i];
  if (threadIdx.x < kC8)               sb[threadIdx.x] = bf[threadIdx.x];
  else if (threadIdx.x < 2 * kC8)      sb[threadIdx.x] = bg[threadIdx.x - kC8];
  else if (threadIdx.x < 2 * kC8 + kC) sb[threadIdx.x] = bh[threadIdx.x - 2 * kC8];
  __syncthreads();

  const int n = (blockIdx.x & 31) * 256 + threadIdx.x;

  float fa[kC8], ga[kC8], va[kC];
#pragma unroll
  for (int k = 0; k < kC8; ++k) { fa[k] = sb[k]; ga[k] = sb[kC8 + k]; }
#pragma unroll
  for (int o = 0; o < kC; ++o) va[o] = sb[2 * kC8 + o];

  for (int c = 0; c < kC; ++c) {
    const float xv = x[c * kN + n];
#pragma unroll
    for (int k = 0; k < kC8; ++k) {
      fa[k] = fmaf(sWf[k * kC + c], xv, fa[k]);
      ga[k] = fmaf(sWg[k * kC + c], xv, ga[k]);
    }
#pragma unroll
    for (int o = 0; o < kC; ++o) va[o] = fmaf(sWh[o * kC + c], xv, va[o]);
  }

  float* wsi = ws + (size_t)img * IMG_STRIDE;
#pragma unroll
  for (int k = 0; k < kC8; ++k) {
    wsi[F_OFF + n * kC8 + k] = fa[k];
    wsi[G_OFF + k * kN + n]  = ga[k];
  }
#pragma unroll
  for (int o = 0; o < kC; ++o) wsi[V_OFF + o * kN + n] = va[o];
}

// ---------------------------------------------------------------------------
// Kernel 2: softmax denominators Z[n] = sum_m exp(l[n,m]),
// l = f @ g with V_WMMA_F32_16X16X4_F32 (two K=4 chunks), f frags hoisted.
// Reduction scratch is wave-private: DS is in-order per wave, so only a
// compiler fence (wave_barrier) is needed, no workgroup barrier.
// grid = 256 blocks (128 per image) x 128 threads  -> 1024 waves
// ---------------------------------------------------------------------------
__global__ __launch_bounds__(128) void rowsum_kernel(float* __restrict__ ws)
{
  const int wid  = threadIdx.x >> 5;
  const int lane = threadIdx.x & 31;
  const int col  = lane & 15;
  const int hi   = lane >> 4;
  const int img  = blockIdx.x >> 7;
  const int nb   = blockIdx.x & 127;

  float* wsi = ws + (size_t)img * IMG_STRIDE;
  const float* f = wsi + F_OFF;
  const float* g = wsi + G_OFF;

  const int n0   = nb * 64 + wid * 16;
  const int nrow = n0 + col;
  // A (f) layout: lanes 0-15 -> K={0,1}, lanes 16-31 -> K={2,3}
  const v2f a0 = *(const v2f*)(f + nrow * kC8 + 2 * hi);
  const v2f a1 = *(const v2f*)(f + nrow * kC8 + 4 + 2 * hi);

  v8f acc = {};
  for (int mt = 0; mt < kN; mt += 16) {
    const int m = mt + col;
    v2f b0, b1;
    b0.x = g[(0 + 2 * hi) * kN + m];
    b0.y = g[(1 + 2 * hi) * kN + m];
    b1.x = g[(4 + 2 * hi) * kN + m];
    b1.y = g[(5 + 2 * hi) * kN + m];
    v8f z = {};
    v8f l = wmma_f32(a0, b0, z);
    l = wmma_f32(a1, b1, l);
#pragma unroll
    for (int r = 0; r < 8; ++r)
      acc[r] += __builtin_amdgcn_exp2f(l[r] * kLog2e);
  }

  // D element (vgpr r, lane) = row (r + 8*hi), col (lane&15): reduce over cols
  __shared__ float red[4][16][16];
#pragma unroll
  for (int r = 0; r < 8; ++r) red[wid][r + 8 * hi][col] = acc[r];
  __builtin_amdgcn_wave_barrier();
  if (lane < 16) {
    float s = 0.f;
#pragma unroll
    for (int t = 0; t < 16; ++t) s += red[wid][lane][t];
    wsi[Z_OFF + n0 + lane] = s;
  }
}

// ---------------------------------------------------------------------------
// Kernel 3: fold softmax denominator and gamma into values:
//           v'[c,n] = gamma * v[c,n] / Z[n]
// ---------------------------------------------------------------------------
__global__ __launch_bounds__(256) void vprime_kernel(
    float* __restrict__ ws,
    const float* __restrict__ gamma1, const float* __restrict__ gamma2)
{
  const int idx = blockIdx.x * 256 + threadIdx.x;   // 2*C*N threads
  const int img = idx / (kC * kN);
  const int rem = idx - img * (kC * kN);
  const int n   = rem & (kN - 1);
  const float gamma = img ? gamma2[0] : gamma1[0];
  float* wsi = ws + (size_t)img * IMG_STRIDE;
  const float z = wsi[Z_OFF + n];
  wsi[V_OFF + rem] = wsi[V_OFF + rem] * (gamma * __builtin_amdgcn_rcpf(z));
}

// ---------------------------------------------------------------------------
// Kernel 4: y[c,m] = sum_n v'[c,n] * exp(l[n,m]) + x[c,m]    (gamma in v')
// WG = 128 threads (4 waves), 64-wide m block, one 16-wide m-subtile per wave.
// Per 16-deep n-tile:
//   - v'[64][16] tile is DMA'd into double-buffered LDS with
//     global_load_async_to_lds_b128 (issued one tile ahead, ASYNCcnt-tracked)
//   - logits E = exp(f@g) via WMMA, staged through wave-private LDS
//     (D->B layout; DS is in-order per wave, no workgroup barrier needed)
//   - acc += v'(16c x 4n) @ E(4n x 16m): 4 K-chunks x 4 c-tiles of WMMA
// One __syncthreads per iteration; fill of tile i+1 overlaps compute of i.
// grid = 256 blocks (128 per image) x 128 threads  -> 1024 waves
// ---------------------------------------------------------------------------
__global__ __launch_bounds__(128) void sa_kernel(
    const float* __restrict__ x1, const float* __restrict__ x2,
    const float* __restrict__ ws, float* __restrict__ out)
{
  const int wid  = threadIdx.x >> 5;
  const int lane = threadIdx.x & 31;
  const int col  = lane & 15;
  const int hi   = lane >> 4;
  const int img  = blockIdx.x >> 7;
  const int mb   = blockIdx.x & 127;

  const float* wsi = ws + (size_t)img * IMG_STRIDE;
  const float* f  = wsi + F_OFF;
  const float* g  = wsi + G_OFF;
  const float* vp = wsi + V_OFF;
  const float* x  = img ? x2 : x1;
  float* y = out + (size_t)img * (kC * kN);

  const int m = mb * 64 + wid * 16 + col;

  // g B-fragments are loop-invariant
  v2f bg0, bg1;
  bg0.x = g[(0 + 2 * hi) * kN + m]; bg0.y = g[(1 + 2 * hi) * kN + m];
  bg1.x = g[(4 + 2 * hi) * kN + m]; bg1.y = g[(5 + 2 * hi) * kN + m];

  v8f zz = {};
  v8f acc[4];
#pragma unroll
  for (int ct = 0; ct < 4; ++ct) acc[ct] = zz;

  __shared__ float E[4][16][17];      // wave-private padded exp tile
  __shared__ float vpt[2][kC][20];    // double-buffered v' tile; stride 20
                                      // floats: 16B-aligned b128 DMA writes,
                                      // 8B-aligned conflict-free b64 reads

  // cooperative-fill indices: thread -> (row, 8-float half-row)
  const int frow  = threadIdx.x >> 1;
  const int fhalf = (threadIdx.x & 1) * 8;
  const float* fill_src = vp + frow * kN + fhalf;

  // prologue: DMA tile 0 into buffer 0
  async_fill_b128(fill_src + 0, &vpt[0][frow][fhalf]);
  async_fill_b128(fill_src + 4, &vpt[0][frow][fhalf + 4]);

  int buf = 0;
  for (int n0 = 0; n0 < kN; n0 += 16) {
    // ---- logits + exp for this wave's m-subtile (independent of vpt) ----
    const int nrow = n0 + col;
    const v2f a0 = *(const v2f*)(f + nrow * kC8 + 2 * hi);
    const v2f a1 = *(const v2f*)(f + nrow * kC8 + 4 + 2 * hi);
    v8f l = wmma_f32(a0, bg0, zz);
    l = wmma_f32(a1, bg1, l);
    // D-layout store: vgpr r -> row r+8*hi, lanes -> cols (wave-private)
#pragma unroll
    for (int r = 0; r < 8; ++r)
      E[wid][r + 8 * hi][col] = __builtin_amdgcn_exp2f(l[r] * kLog2e);

    // ---- own DMA done -> barrier -> all waves' fills of `buf` visible;
    //      barrier also fences last iteration's reads of buf^1 ----
    wait_async0();
    __syncthreads();

    // ---- prefetch next tile into the other buffer (uniform branch) ----
    if (n0 + 16 < kN) {
      const float* s = fill_src + n0 + 16;
      async_fill_b128(s + 0, &vpt[buf ^ 1][frow][fhalf]);
      async_fill_b128(s + 4, &vpt[buf ^ 1][frow][fhalf + 4]);
    }
    __builtin_amdgcn_wave_barrier();

    // ---- acc += v'(16c x 4n) @ E(4n x 16m), 4 K-chunks x 4 c-tiles ----
#pragma unroll
    for (int q = 0; q < 4; ++q) {
      const int k0 = 4 * q + 2 * hi;  // B layout: lanes 0-15 K={k0,k0+1} ...
      v2f be;
      be.x = E[wid][k0][col];
      be.y = E[wid][k0 + 1][col];
#pragma unroll
      for (int ct = 0; ct < 4; ++ct) {
        const v2f av = *(const v2f*)(&vpt[buf][ct * 16 + col][k0]);
        acc[ct] = wmma_f32(av, be, acc[ct]);
      }
    }
    __builtin_amdgcn_wave_barrier();
    buf ^= 1;
  }

#pragma unroll
  for (int ct = 0; ct < 4; ++ct) {
#pragma unroll
    for (int r = 0; r < 8; ++r) {
      const int c = ct * 16 + r + 8 * hi;
      y[c * kN + m] = acc[ct][r] + x[c * kN + m];
    }
  }
}

// ---------------------------------------------------------------------------
extern "C" void kernel_launch(void* const* d_in, const int* in_sizes, int n_in,
                              void* d_out, int out_size, void* d_ws, size_t ws_size,
                              hipStream_t stream)
{
  (void)in_sizes; (void)n_in; (void)out_size; (void)ws_size;
  const float* x1  = (const float*)d_in[0];
  const float* x2  = (const float*)d_in[1];
  const float* Wf  = (const float*)d_in[2];
  const float* bf  = (const float*)d_in[3];
  const float* Wg  = (const float*)d_in[4];
  const float* bg  = (const float*)d_in[5];
  const float* Wh1 = (const float*)d_in[6];
  const float* bh1 = (const float*)d_in[7];
  const float* Wh2 = (const float*)d_in[8];
  const float* bh2 = (const float*)d_in[9];
  const float* g1  = (const float*)d_in[10];
  const float* g2  = (const float*)d_in[11];
  float* out = (float*)d_out;
  float* ws  = (float*)d_ws;   // needs 2*IMG_STRIDE*4 bytes ~= 5.1 MB

  proj_kernel<<<64, 256, 0, stream>>>(x1, x2, Wf, bf, Wg, bg,
                                      Wh1, bh1, Wh2, bh2, ws);
  rowsum_kernel<<<256, 128, 0, stream>>>(ws);
  vprime_kernel<<<4096, 256, 0, stream>>>(ws, g1, g2);
  sa_kernel<<<256, 128, 0, stream>>>(x1, x2, ws, out);
}